// CantileverPINN_42167988912355
// MI455X (gfx1250) — compile-verified
//
#include <hip/hip_runtime.h>
#include <hip/hip_bf16.h>

typedef float v2f __attribute__((ext_vector_type(2)));
typedef float v8f __attribute__((ext_vector_type(8)));

#define WAVES 4
#define PTS_PER_WAVE 16
#define PTS_PER_BLOCK (WAVES * PTS_PER_WAVE)

static __device__ __forceinline__ v8f wmma_f32(v2f a, v2f b, v8f c) {
  // V_WMMA_F32_16X16X4_F32 : D(16x16,f32) = A(16x4,f32) * B(4x16,f32) + C
  return __builtin_amdgcn_wmma_f32_16x16x4_f32(
      /*neg_a=*/false, a, /*neg_b=*/false, b,
      /*c_mod=*/(short)0, c, /*reuse_a=*/false, /*reuse_b=*/false);
}

// Hardware V_TANH_F32 (gfx1250 TRANS op) — falls back to OCML if builtin absent.
static __device__ __forceinline__ float fast_tanh(float x) {
#if __has_builtin(__builtin_amdgcn_tanhf)
  return __builtin_amdgcn_tanhf(x);
#elif __has_builtin(__builtin_amdgcn_tanh_f32)
  return __builtin_amdgcn_tanh_f32(x);
#else
  return tanhf(x);
#endif
}

// tanh derivative factors: t=tanh(z); f1=1-t^2; f2=-2 t f1; f3=f1(6t^2-2); f4=8 t f1 (2-3t^2)
__global__ __launch_bounds__(128) void pinn_beam_w4_kernel(
    const float* __restrict__ X,
    const float* __restrict__ W1, const float* __restrict__ B1,
    const float* __restrict__ W2, const float* __restrict__ B2,
    const float* __restrict__ W3, const float* __restrict__ B3,
    const float* __restrict__ W4,
    float* __restrict__ partials, int n)
{
  __shared__ float sW1[16], sB1[16];
  __shared__ float sW2[16 * 32];   // padded [K=16][N=32]
  __shared__ float sB2[32];
  __shared__ float sW3[32 * 64];   // padded [K=32][N=64]
  __shared__ float sB3[64];
  __shared__ float sW4[64];
  __shared__ float sH[WAVES][5][512];   // per-wave jet activations (reused L1 then L2)
  __shared__ float sPart[WAVES];

  const int tid  = threadIdx.x;
  const int lane = tid & 31;
  const int wv   = tid >> 5;
  const int m    = lane & 15;      // point row (A/C) or column (B/C)
  const int hi   = lane >> 4;      // lane-half select
  const int kk   = hi << 1;        // K sub-offset for A/B f32 layout

  // ---- stage zero-padded weights into LDS ----
  for (int i = tid; i < 16 * 32; i += blockDim.x) {
    int k = i >> 5, nn = i & 31;
    sW2[i] = (k < 15 && nn < 30) ? W2[k * 30 + nn] : 0.f;
  }
  for (int i = tid; i < 32 * 64; i += blockDim.x) {
    int k = i >> 6, nn = i & 63;
    sW3[i] = (k < 30 && nn < 60) ? W3[k * 60 + nn] : 0.f;
  }
  for (int i = tid; i < 64; i += blockDim.x) {
    sB3[i] = (i < 60) ? B3[i] : 0.f;
    sW4[i] = (i < 60) ? W4[i] : 0.f;
    if (i < 32) sB2[i] = (i < 30) ? B2[i] : 0.f;
    if (i < 16) { sW1[i] = (i < 15) ? W1[i] : 0.f; sB1[i] = (i < 15) ? B1[i] : 0.f; }
  }
  __syncthreads();

  const int tileBase = (blockIdx.x * WAVES + wv) * PTS_PER_WAVE;

  // ---- Layer 1 (1 -> 15): elementwise jet; seed = (x, 1, 0, 0, 0) ----
  // H1 layout: sH[wv][ch][p*16 + nn], nn padded to 16
  for (int idx = lane; idx < 256; idx += 32) {
    int p = idx >> 4, nn = idx & 15;
    int g = tileBase + p;
    float xv = X[g < n ? g : (n - 1)];
    float w  = sW1[nn];
    float z0 = xv * w + sB1[nn];
    float t  = fast_tanh(z0);
    float t2 = t * t, f1 = 1.f - t2;
    float f2 = -2.f * t * f1;
    float f3 = f1 * (6.f * t2 - 2.f);
    float f4 = 8.f * t * f1 * (2.f - 3.f * t2);
    float w2 = w * w;
    sH[wv][0][p * 16 + nn] = t;
    sH[wv][1][p * 16 + nn] = f1 * w;
    sH[wv][2][p * 16 + nn] = f2 * w2;
    sH[wv][3][p * 16 + nn] = f3 * w2 * w;
    sH[wv][4][p * 16 + nn] = f4 * w2 * w2;
  }

  const v8f vzero = {0.f, 0.f, 0.f, 0.f, 0.f, 0.f, 0.f, 0.f};

  // ---- Layer 2 (15 -> 30), WMMA per channel; keep both N-tiles in regs ----
  v8f acc2[2][5];
  #pragma unroll
  for (int nt = 0; nt < 2; ++nt) {
    #pragma unroll
    for (int ch = 0; ch < 5; ++ch) acc2[nt][ch] = vzero;
    #pragma unroll
    for (int c = 0; c < 16; c += 4) {
      v2f b;
      b.x = sW2[(c + kk) * 32 + nt * 16 + m];
      b.y = sW2[(c + kk + 1) * 32 + nt * 16 + m];
      #pragma unroll
      for (int ch = 0; ch < 5; ++ch) {
        v2f a;
        a.x = sH[wv][ch][m * 16 + c + kk];
        a.y = sH[wv][ch][m * 16 + c + kk + 1];
        acc2[nt][ch] = wmma_f32(a, b, acc2[nt][ch]);
      }
    }
    // bias only on value channel; then jet-tanh in place (lane-local)
    float bb = sB2[nt * 16 + m];
    #pragma unroll
    for (int e = 0; e < 8; ++e) {
      float z0 = acc2[nt][0][e] + bb;
      float z1 = acc2[nt][1][e], z2 = acc2[nt][2][e];
      float z3 = acc2[nt][3][e], z4 = acc2[nt][4][e];
      float t  = fast_tanh(z0);
      float t2 = t * t, f1 = 1.f - t2;
      float f2 = -2.f * t * f1;
      float f3 = f1 * (6.f * t2 - 2.f);
      float f4 = 8.f * t * f1 * (2.f - 3.f * t2);
      acc2[nt][0][e] = t;
      acc2[nt][1][e] = f1 * z1;
      acc2[nt][2][e] = f2 * z1 * z1 + f1 * z2;
      acc2[nt][3][e] = f3 * z1 * z1 * z1 + 3.f * f2 * z1 * z2 + f1 * z3;
      acc2[nt][4][e] = f4 * z1 * z1 * z1 * z1 + 6.f * f3 * z1 * z1 * z2
                     + f2 * (3.f * z2 * z2 + 4.f * z1 * z3) + f1 * z4;
    }
  }
  // store H2 jets (C-layout -> [p][n], stride 32); all L2 GEMM reads are done
  #pragma unroll
  for (int nt = 0; nt < 2; ++nt) {
    #pragma unroll
    for (int e = 0; e < 8; ++e) {
      int p  = e + (hi << 3);
      int nn = nt * 16 + m;
      #pragma unroll
      for (int ch = 0; ch < 5; ++ch)
        sH[wv][ch][p * 32 + nn] = acc2[nt][ch][e];
    }
  }

  // ---- Layer 3 (30 -> 60) fused with Layer 4 (60 -> 1, channel-4 only) ----
  float s[8];
  #pragma unroll
  for (int e = 0; e < 8; ++e) s[e] = 0.f;

  #pragma unroll
  for (int nt = 0; nt < 4; ++nt) {
    v8f acc[5];
    #pragma unroll
    for (int ch = 0; ch < 5; ++ch) acc[ch] = vzero;
    #pragma unroll
    for (int c = 0; c < 32; c += 4) {
      v2f b;
      b.x = sW3[(c + kk) * 64 + nt * 16 + m];
      b.y = sW3[(c + kk + 1) * 64 + nt * 16 + m];
      #pragma unroll
      for (int ch = 0; ch < 5; ++ch) {
        v2f a;
        a.x = sH[wv][ch][m * 32 + c + kk];
        a.y = sH[wv][ch][m * 32 + c + kk + 1];
        acc[ch] = wmma_f32(a, b, acc[ch]);
      }
    }
    float bb  = sB3[nt * 16 + m];
    float w4l = sW4[nt * 16 + m];   // zero-padded beyond n=60
    #pragma unroll
    for (int e = 0; e < 8; ++e) {
      float z0 = acc[0][e] + bb;
      float z1 = acc[1][e], z2 = acc[2][e], z3 = acc[3][e], z4 = acc[4][e];
      float t  = fast_tanh(z0);
      float t2 = t * t, f1 = 1.f - t2;
      float f2 = -2.f * t * f1;
      float f3 = f1 * (6.f * t2 - 2.f);
      float f4 = 8.f * t * f1 * (2.f - 3.f * t2);
      float y4 = f4 * z1 * z1 * z1 * z1 + 6.f * f3 * z1 * z1 * z2
               + f2 * (3.f * z2 * z2 + 4.f * z1 * z3) + f1 * z4;
      s[e] += y4 * w4l;             // partial of w'''' over this lane's neuron
    }
  }

  // reduce w'''' across the 16-lane halves (N dimension lives on lanes)
  #pragma unroll
  for (int off = 1; off <= 8; off <<= 1) {
    #pragma unroll
    for (int e = 0; e < 8; ++e) s[e] += __shfl_xor(s[e], off, 32);
  }
  // half 0 holds points e, half 1 holds points e+8; residual = w'''' - P/(E*I) = w'''' - 1
  float ssq = 0.f;
  #pragma unroll
  for (int e = 0; e < 8; ++e) {
    int p = tileBase + e + (hi << 3);
    float r = s[e] - 1.0f;
    ssq += (p < n) ? r * r : 0.f;
  }
  ssq += __shfl_xor(ssq, 16, 32);
  if (lane == 0) sPart[wv] = ssq;
  __syncthreads();
  if (tid == 0) {
    float tot = 0.f;
    for (int w = 0; w < WAVES; ++w) tot += sPart[w];
    partials[blockIdx.x] = tot;
  }
}

__global__ __launch_bounds__(256) void pinn_beam_reduce_kernel(
    const float* __restrict__ partials, float* __restrict__ out,
    int nparts, float inv_n)
{
  __shared__ float sm[256];
  float s = 0.f;
  for (int i = threadIdx.x; i < nparts; i += 256) s += partials[i];
  sm[threadIdx.x] = s;
  __syncthreads();
  for (int off = 128; off > 0; off >>= 1) {
    if (threadIdx.x < off) sm[threadIdx.x] += sm[threadIdx.x + off];
    __syncthreads();
  }
  if (threadIdx.x == 0) out[0] = sm[0] * inv_n;
}

extern "C" void kernel_launch(void* const* d_in, const int* in_sizes, int n_in,
                              void* d_out, int out_size, void* d_ws, size_t ws_size,
                              hipStream_t stream) {
  const float* X  = (const float*)d_in[0];
  const float* W1 = (const float*)d_in[1];
  const float* B1 = (const float*)d_in[2];
  const float* W2 = (const float*)d_in[3];
  const float* B2 = (const float*)d_in[4];
  const float* W3 = (const float*)d_in[5];
  const float* B3 = (const float*)d_in[6];
  const float* W4 = (const float*)d_in[7];
  // d_in[8] = b4: does not affect the 4th derivative -> unused

  const int n = in_sizes[0];
  const int blocks = (n + PTS_PER_BLOCK - 1) / PTS_PER_BLOCK;  // 4096 for N=262144
  float* partials = (float*)d_ws;                               // 4 B * blocks scratch

  pinn_beam_w4_kernel<<<blocks, 128, 0, stream>>>(X, W1, B1, W2, B2, W3, B3, W4,
                                                  partials, n);
  pinn_beam_reduce_kernel<<<1, 256, 0, stream>>>(partials, (float*)d_out, blocks,
                                                 1.0f / (float)n);
}